// ChamferDistance_26963804685126
// MI455X (gfx1250) — compile-verified
//
#include <hip/hip_runtime.h>
#include <hip/hip_bf16.h>

// Chamfer distance via V_WMMA_F32_16X16X4_F32 (K=4 f32 matrix pipe).
// D[i,j] = t2_j - 2 p_i.t_j computed as a 16x16x4 GEMM tile:
//   A[i,:] = [-2px, -2py, -2pz, 1]   (query rows)
//   B[:,j] = [tx, ty, tz, t2_j]      (reference cols)
// d[i,j] = p2_i + D[i,j]; row-min accumulated elementwise in C/D-layout
// registers across the whole j sweep (adding the row-constant p2_i is
// order-preserving, so it is folded in after the final reduction).
// Both chamfer directions are symmetric row-min problems -> one kernel,
// blockIdx.z selects direction (doubles concurrent waves vs two launches).

typedef __attribute__((ext_vector_type(2))) float v2f;
typedef __attribute__((ext_vector_type(8))) float v8f;

#define WAVES_PER_WG 8   // 256 threads = 8 wave32; each wave owns a 16-row tile

__global__ __launch_bounds__(256) void chamfer_rowmin_wmma(
    const float* __restrict__ pred,    // [B, N, 3]
    const float* __restrict__ target,  // [B, M, 3]
    float* __restrict__ chamx,         // [B, N] min over target of ||p-t||^2
    float* __restrict__ chamy,         // [B, M] min over pred   of ||t-p||^2
    int N, int M)
{
  // Direction select: z==0 -> rows from pred, cols from target -> chamx
  //                   z==1 -> rows from target, cols from pred -> chamy
  const int dir = blockIdx.z;
  const float* P = dir ? target : pred;
  const float* T = dir ? pred : target;
  float* rowmin  = dir ? chamy : chamx;
  const int NP   = dir ? M : N;
  const int NT   = dir ? N : M;

  const int lane = threadIdx.x & 31;
  const int wave = threadIdx.x >> 5;
  const int b    = blockIdx.y;
  const int i0   = (blockIdx.x * WAVES_PER_WG + wave) * 16;
  if (i0 >= NP) return;  // uniform per wave; EXEC stays all-ones for WMMA

  const float* Pb = P + (size_t)b * NP * 3;
  const float* Tb = T + (size_t)b * NT * 3;

  const int h = lane >> 4;          // lane half (selects K pair of fragment)
  const int m = i0 + (lane & 15);   // row this lane contributes to A

  // Load this lane's query point (both halves read the same row).
  const float px = Pb[m * 3 + 0];
  const float py = Pb[m * 3 + 1];
  const float pz = Pb[m * 3 + 2];
  const float p2own = px * px + py * py + pz * pz;

  // A fragment (16x4 f32 layout: half0 -> K={0,1}, half1 -> K={2,3}).
  v2f a;
  a.x = h ? (-2.0f * pz) : (-2.0f * px);
  a.y = h ? 1.0f         : (-2.0f * py);

  // p2 for the row owned by accumulator slot r: row = i0 + r + 8*h.
  float p2reg[8];
#pragma unroll
  for (int r = 0; r < 8; ++r) p2reg[r] = __shfl(p2own, r + 8 * h, 32);

  const float INF = __builtin_huge_valf();
  v8f acc;
#pragma unroll
  for (int r = 0; r < 8; ++r) acc[r] = INF;

  // Sweep all NT reference points in 16-col tiles:
  // 1 b96 load + ~6 prep VALU + 1 WMMA + (min3-fused) 4 min per tile.
#pragma unroll 2
  for (int jt = 0; jt < NT; jt += 16) {
    const int j = jt + (lane & 15);
    const float tx = Tb[j * 3 + 0];
    const float ty = Tb[j * 3 + 1];
    const float tz = Tb[j * 3 + 2];
    const float t2 = tx * tx + ty * ty + tz * tz;

    // B fragment (4x16): half0 -> K={0,1} = (tx,ty); half1 -> K={2,3} = (tz,t2).
    v2f bf;
    bf.x = h ? tz : tx;
    bf.y = h ? t2 : ty;

    v8f zero = {};
    v8f d = __builtin_amdgcn_wmma_f32_16x16x4_f32(
        /*neg_a=*/false, a, /*neg_b=*/false, bf,
        /*c_mod=*/(short)0, zero, /*reuse_a=*/false, /*reuse_b=*/false);

#pragma unroll
    for (int r = 0; r < 8; ++r) acc[r] = fminf(acc[r], d[r]);
  }

  // Min-reduce each acc slot across the 16 lanes of its half (xor<16 stays in-half).
#pragma unroll
  for (int s = 1; s < 16; s <<= 1) {
#pragma unroll
    for (int r = 0; r < 8; ++r) {
      float v = acc[r];
      acc[r] = fminf(v, __shfl_xor(v, s, 32));
    }
  }

  // Lane 0 writes rows i0..i0+7, lane 16 writes rows i0+8..i0+15.
  if ((lane & 15) == 0) {
#pragma unroll
    for (int r = 0; r < 8; ++r)
      rowmin[(size_t)b * NP + i0 + r + 8 * h] = acc[r] + p2reg[r];
  }
}

// Final reduction: loss = sum(chamx)/(B*N) + sum(chamy)/(B*M).
__global__ __launch_bounds__(256) void chamfer_reduce(
    const float* __restrict__ cham, int nx, int ny, float sx, float sy,
    float* __restrict__ out)
{
  __shared__ float sdata[256];
  float s = 0.0f;
  for (int i = threadIdx.x; i < nx; i += 256) s += cham[i] * sx;
  for (int i = threadIdx.x; i < ny; i += 256) s += cham[nx + i] * sy;
  sdata[threadIdx.x] = s;
  __syncthreads();
  for (int t = 128; t > 0; t >>= 1) {
    if (threadIdx.x < t) sdata[threadIdx.x] += sdata[threadIdx.x + t];
    __syncthreads();
  }
  if (threadIdx.x == 0) out[0] = sdata[0];
}

extern "C" void kernel_launch(void* const* d_in, const int* in_sizes, int n_in,
                              void* d_out, int out_size, void* d_ws, size_t ws_size,
                              hipStream_t stream) {
  (void)in_sizes; (void)n_in; (void)out_size; (void)ws_size;
  const float* pred   = (const float*)d_in[0];
  const float* target = (const float*)d_in[1];

  // Shapes fixed by the reference: [16, 4096, 3] both sides.
  const int B = 16, N = 4096, M = 4096;

  float* chamx = (float*)d_ws;               // [B*N]
  float* chamy = chamx + (size_t)B * N;      // [B*M]

  // One launch covers both directions (z): 2 * B * N/16 = 8192 waves in flight.
  dim3 blk(256);
  dim3 grd((N + 16 * WAVES_PER_WG - 1) / (16 * WAVES_PER_WG), B, 2);
  chamfer_rowmin_wmma<<<grd, blk, 0, stream>>>(pred, target, chamx, chamy, N, M);

  chamfer_reduce<<<1, 256, 0, stream>>>(chamx, B * N, B * M,
                                        1.0f / (float)(B * N),
                                        1.0f / (float)(B * M),
                                        (float*)d_out);
}